// Capsule_4071628997246
// MI455X (gfx1250) — compile-verified
//
#include <hip/hip_runtime.h>
#include <hip/hip_bf16.h>

typedef __attribute__((ext_vector_type(2))) float v2f;
typedef __attribute__((ext_vector_type(8))) float v8f;

static __device__ __forceinline__ v8f wmma_f32_k4(v2f a, v2f b, v8f c) {
  // D = A(16x4) * B(4x16) + C, fp32 throughout
  return __builtin_amdgcn_wmma_f32_16x16x4_f32(false, a, false, b, (short)0, c, false, false);
}
static __device__ __forceinline__ float sigm(float x) { return 1.0f / (1.0f + __expf(-x)); }

// ---------------------------------------------------------------- embedding
__global__ void k_embed(const int* __restrict__ idx, const float* __restrict__ embed,
                        float* __restrict__ enc) {
  int bt = blockIdx.x;            // 0..479  (b*30+t)
  int e  = threadIdx.x;           // 0..255
  int tok = idx[bt];
  enc[(size_t)bt * 256 + e] = embed[(size_t)tok * 256 + e];
}

// ------------------------------------------- LSTM input-side GEMM (WMMA f32)
// xg[dir][t*16+b][n] = X[b][t][:] . Wih[dir][n][:] + bih + bhh
__global__ void k_xg(const float* __restrict__ X, const float* __restrict__ Wih,
                     const float* __restrict__ bih, const float* __restrict__ bhh,
                     float* __restrict__ xg, int K) {
  int dir  = blockIdx.z;
  int mt   = blockIdx.x;                    // 30 M-tiles of 16 rows (m = t*16+b)
  int wave = threadIdx.x >> 5;
  int lane = threadIdx.x & 31;
  int hi   = lane >> 4;
  int ncol = blockIdx.y * 128 + wave * 16 + (lane & 15);   // 0..1023
  int mrow = mt * 16 + (lane & 15);
  int t = mrow >> 4, b = mrow & 15;
  const float* Arow = X + ((size_t)b * 30 + t) * K;
  const float* Bcol = Wih + (size_t)dir * 1024 * K + (size_t)ncol * K;
  v8f acc = {0.f,0.f,0.f,0.f,0.f,0.f,0.f,0.f};
  for (int k = 0; k < K; k += 4) {
    int kb = k + 2 * hi;
    v2f a;  a.x  = Arow[kb]; a.y  = Arow[kb + 1];
    v2f bv; bv.x = Bcol[kb]; bv.y = Bcol[kb + 1];
    acc = wmma_f32_k4(a, bv, acc);
  }
  float bias = bih[dir * 1024 + ncol] + bhh[dir * 1024 + ncol];
  float* outb = xg + (size_t)dir * 480 * 1024;
  for (int v = 0; v < 8; v++) {
    int row = mt * 16 + v + 8 * hi;
    outb[(size_t)row * 1024 + ncol] = acc[v] + bias;
  }
}

// ----------------------------------------- recurrent LSTM (persistent, WMMA)
// 2 blocks (fwd/rev) x 512 threads (16 waves). Wave w owns hidden units
// j0=16w..16w+15 and computes all 4 gate tiles; cell state in registers.
__global__ void k_lstm_rec(const float* __restrict__ xg, const float* __restrict__ Whh,
                           float* __restrict__ hout) {
  __shared__ float sh[16 * 256];            // h state [b][j]
  int dir  = blockIdx.x;
  int tid  = threadIdx.x;
  int wave = tid >> 5, lane = tid & 31;
  int hi   = lane >> 4;
  int jc   = wave * 16 + (lane & 15);       // hidden unit column 0..255
  int m    = lane & 15;                     // A row for LDS load
  for (int i = tid; i < 4096; i += 512) sh[i] = 0.f;
  float creg[8];
  for (int v = 0; v < 8; v++) creg[v] = 0.f;
  __syncthreads();
  const float* xgd = xg + (size_t)dir * 480 * 1024;
  const float* Wd  = Whh + (size_t)dir * 1024 * 256;
  const float* Bi = Wd + (size_t)(0 * 256 + jc) * 256;
  const float* Bf = Wd + (size_t)(1 * 256 + jc) * 256;
  const float* Bg = Wd + (size_t)(2 * 256 + jc) * 256;
  const float* Bo = Wd + (size_t)(3 * 256 + jc) * 256;
  for (int s = 0; s < 30; s++) {
    int t = dir ? (29 - s) : s;
    const float* xrow = xgd + (size_t)t * 16 * 1024;
    v8f ai = {0.f,0.f,0.f,0.f,0.f,0.f,0.f,0.f};
    v8f af = ai, ag = ai, ao = ai;
    for (int v = 0; v < 8; v++) {
      const float* xr = xrow + (size_t)(v + 8 * hi) * 1024;
      ai[v] = xr[0 * 256 + jc];
      af[v] = xr[1 * 256 + jc];
      ag[v] = xr[2 * 256 + jc];
      ao[v] = xr[3 * 256 + jc];
    }
    for (int k = 0; k < 256; k += 4) {
      int kb = k + 2 * hi;
      v2f a; a.x = sh[m * 256 + kb]; a.y = sh[m * 256 + kb + 1];
      v2f bv;
      bv.x = Bi[kb]; bv.y = Bi[kb + 1]; ai = wmma_f32_k4(a, bv, ai);
      bv.x = Bf[kb]; bv.y = Bf[kb + 1]; af = wmma_f32_k4(a, bv, af);
      bv.x = Bg[kb]; bv.y = Bg[kb + 1]; ag = wmma_f32_k4(a, bv, ag);
      bv.x = Bo[kb]; bv.y = Bo[kb + 1]; ao = wmma_f32_k4(a, bv, ao);
    }
    float hnew[8];
    for (int v = 0; v < 8; v++) {
      float cc = sigm(af[v]) * creg[v] + sigm(ai[v]) * tanhf(ag[v]);
      hnew[v] = sigm(ao[v]) * tanhf(cc);
      creg[v] = cc;
    }
    __syncthreads();                         // all reads of sh done
    for (int v = 0; v < 8; v++) {
      int b = v + 8 * hi;
      sh[b * 256 + jc] = hnew[v];
      hout[((size_t)b * 30 + t) * 512 + dir * 256 + jc] = hnew[v];
    }
    __syncthreads();
  }
}

// ----------------------------------------------------------------------- fc
__global__ void k_fc(const float* __restrict__ h, const float* __restrict__ W,
                     const float* __restrict__ bias, float* __restrict__ out) {
  int i = blockIdx.x * blockDim.x + threadIdx.x;
  if (i >= 16 * 30 * 30) return;
  int tt = i % 30, bt = i / 30;
  const float* hr = h + (size_t)bt * 512;
  const float* wr = W + (size_t)tt * 512;
  float s = bias[tt];
  for (int j = 0; j < 512; j++) s += hr[j] * wr[j];
  out[i] = s;
}

// ------------------------------------------------------ conv1 7x7 + relu
__global__ void k_conv1(const float* __restrict__ in, const float* __restrict__ W,
                        const float* __restrict__ bias, float* __restrict__ out) {
  int i = blockIdx.x * blockDim.x + threadIdx.x;
  if (i >= 16 * 256 * 576) return;
  int px = i % 576, co = (i / 576) & 255, b = i / (576 * 256);
  int oy = px / 24, ox = px % 24;
  const float* ip = in + (size_t)b * 900 + oy * 30 + ox;
  const float* wp = W + (size_t)co * 49;
  float s = bias[co];
  for (int ky = 0; ky < 7; ky++)
    for (int kx = 0; kx < 7; kx++)
      s += ip[ky * 30 + kx] * wp[ky * 7 + kx];
  out[i] = fmaxf(s, 0.f);
}

// --------------------- primary-capsule conv as implicit GEMM (WMMA f32)
// M=6400 (b,oy,ox), N=256 (co), K=6400 (ci,ky,kx)
// A-tiles (16x128) double-buffered in LDS via CDNA5 async memory->LDS DMA:
// chunk i+1 streams into buffer (i+1)%2 while WMMAs consume buffer i%2.
__global__ void k_pconv(const float* __restrict__ x, const float* __restrict__ W,
                        const float* __restrict__ bias, float* __restrict__ y) {
  __shared__ float sA[2][16 * 132];         // row stride 132: bank-conflict free
  int mt = blockIdx.x;                      // 400 M-tiles
  int ng = blockIdx.y;                      // 2 N-groups of 128
  int tid = threadIdx.x;
  int wave = tid >> 5, lane = tid & 31;
  int hi = lane >> 4;
  int co = ng * 128 + wave * 16 + (lane & 15);
  const float* Bcol = W + (size_t)co * 6400;
  // LDS byte offsets (flat-aperture rule: addr[31:0] == LDS offset)
  unsigned sb0 = (unsigned)(unsigned long long)(const void*)&sA[0][0];
  unsigned sb1 = (unsigned)(unsigned long long)(const void*)&sA[1][0];
  // Per-thread invariant im2col coords for its 8 (row,kk) tile slots.
  const float* gbase[8];
  unsigned     loff[8];
  for (int e = 0; e < 8; e++) {
    int idx = tid + e * 256;                // 0..2047
    int row = idx >> 7, kk = idx & 127;
    int mm = mt * 16 + row;
    int b = mm / 400, pix = mm % 400;
    int oy = pix / 20, ox = pix % 20;
    gbase[e] = x + ((size_t)(b * 256) * 24 + oy) * 24 + ox;  // + ci*576 + ky*24 + kx
    loff[e]  = (unsigned)((row * 132 + kk) * 4);
  }
  // issue one chunk's 8 gather-DMAs into LDS buffer at byte base `sb`
  auto issue = [&](int kc, unsigned sb) {
    for (int e = 0; e < 8; e++) {
      int k = kc + ((tid + e * 256) & 127);
      int ci = k / 25, r = k % 25, ky = r / 5, kx = r % 5;
      const float* gp = gbase[e] + ci * 576 + ky * 24 + kx;
      unsigned lds_off = sb + loff[e];
      asm volatile("global_load_async_to_lds_b32 %0, %1, off"
                   :: "v"(lds_off), "v"(gp) : "memory");
    }
  };
  v8f acc = {0.f,0.f,0.f,0.f,0.f,0.f,0.f,0.f};
  issue(0, sb0);                            // preload chunk 0 -> buf 0
  int m = lane & 15;
  for (int i = 0; i < 50; i++) {
    int kc = i * 128;
    __syncthreads();                        // all waves done reading buf (i+1)%2
    if (i + 1 < 50) {
      issue(kc + 128, (i & 1) ? sb0 : sb1); // chunk i+1 -> buf (i+1)%2
      asm volatile("s_wait_asynccnt 0x8" ::: "memory");  // chunk i complete (in-order)
      __builtin_prefetch(Bcol + kc + 128, 0, 1);
    } else {
      asm volatile("s_wait_asynccnt 0x0" ::: "memory");
    }
    __syncthreads();                        // chunk i visible to all waves
    const float* cur = &sA[i & 1][0];
    for (int ks = 0; ks < 128; ks += 4) {
      int kb = ks + 2 * hi;
      v2f a;  a.x  = cur[m * 132 + kb];  a.y  = cur[m * 132 + kb + 1];
      v2f bv; bv.x = Bcol[kc + kb];      bv.y = Bcol[kc + kb + 1];
      acc = wmma_f32_k4(a, bv, acc);
    }
  }
  float bb = bias[co];
  for (int v = 0; v < 8; v++) {
    int mm = mt * 16 + v + 8 * hi;
    int b = mm / 400, pix = mm % 400;
    y[((size_t)(b * 256 + co)) * 400 + pix] = acc[v] + bb;
  }
}

// -------------------------- squash scale per (b, capsule-unit n): 12800-red
__global__ void k_scale(const float* __restrict__ y, float* __restrict__ scale) {
  __shared__ float red[256];
  int b = blockIdx.x, n = blockIdx.y;
  const float* p = y + ((size_t)b * 256 + n * 32) * 400;
  float s = 0.f;
  for (int i = threadIdx.x; i < 12800; i += 256) { float v = p[i]; s += v * v; }
  red[threadIdx.x] = s; __syncthreads();
  for (int o = 128; o > 0; o >>= 1) {
    if (threadIdx.x < o) red[threadIdx.x] += red[threadIdx.x + o];
    __syncthreads();
  }
  if (threadIdx.x == 0) {
    float msq = red[0];
    scale[b * 8 + n] = msq / (1.0f + msq) * rsqrtf(msq);
  }
}

// ------------------ u_mat[b][p*8+n] = scale[b][n] * y[b][n*12800+p]
__global__ void k_umat(const float* __restrict__ y, const float* __restrict__ scale,
                       float* __restrict__ u) {
  int i = blockIdx.x * blockDim.x + threadIdx.x;
  if (i >= 16 * 102400) return;
  int b = i / 102400, k = i % 102400;
  int p = k >> 3, n = k & 7;
  u[i] = scale[b * 8 + n] * y[(size_t)b * 102400 + n * 12800 + p];
}

__global__ void k_szero(float* __restrict__ s) {
  int i = blockIdx.x * blockDim.x + threadIdx.x;
  if (i < 16 * 160) s[i] = 0.f;
}

// ------------- s[b][l*16+o] = sum_{p,n} W[p][l][o][n] * u[b][p*8+n] (WMMA)
__global__ void k_sgemm(const float* __restrict__ u, const float* __restrict__ W,
                        float* __restrict__ s) {
  int kc0 = blockIdx.x * 1600;              // 64 K-chunks
  int wave = threadIdx.x >> 5, lane = threadIdx.x & 31;   // 10 waves
  int hi = lane >> 4;
  int col = wave * 16 + (lane & 15);        // 0..159 = l*16+o
  int m = lane & 15;                        // batch row
  const float* urow = u + (size_t)m * 102400;
  v8f acc = {0.f,0.f,0.f,0.f,0.f,0.f,0.f,0.f};
  for (int k = kc0; k < kc0 + 1600; k += 4) {
    int kb = k + 2 * hi;
    int p = kb >> 3, n = kb & 7;            // kb even -> n even -> n+1 contiguous
    v2f a;  a.x  = urow[kb]; a.y = urow[kb + 1];
    const float* wp = W + (size_t)p * 1280 + col * 8 + n;
    v2f bv; bv.x = wp[0]; bv.y = wp[1];
    acc = wmma_f32_k4(a, bv, acc);
  }
  for (int v = 0; v < 8; v++) {
    int row = v + 8 * hi;                   // b
    atomicAdd(&s[row * 160 + col], acc[v]);
  }
}

// ------------------------- final squash over L, write v to d_out[0..2559]
__global__ void k_vout(const float* __restrict__ s, float* __restrict__ out) {
  int tid = threadIdx.x;                    // 256 = (b,o)
  int b = tid >> 4, o = tid & 15;
  float msq = 0.f;
  for (int l = 0; l < 10; l++) { float v = s[b * 160 + l * 16 + o]; msq += v * v; }
  float f = msq / (1.0f + msq) * rsqrtf(msq);
  for (int l = 0; l < 10; l++)
    out[(b * 10 + l) * 16 + o] = f * s[b * 160 + l * 16 + o];
}

// ------------------------------- lstm_feats = mean_t h1 -> d_out[2560..]
__global__ void k_feats(const float* __restrict__ h1, float* __restrict__ out) {
  int i = blockIdx.x * blockDim.x + threadIdx.x;
  if (i >= 16 * 512) return;
  int b = i >> 9, j = i & 511;
  float s = 0.f;
  for (int t = 0; t < 30; t++) s += h1[((size_t)b * 30 + t) * 512 + j];
  out[2560 + i] = s * (1.0f / 30.0f);
}

extern "C" void kernel_launch(void* const* d_in, const int* in_sizes, int n_in,
                              void* d_out, int out_size, void* d_ws, size_t ws_size,
                              hipStream_t stream) {
  const int*   input     = (const int*)  d_in[0];
  const float* embed     = (const float*)d_in[1];
  const float* l0_Wih    = (const float*)d_in[2];
  const float* l0_Whh    = (const float*)d_in[3];
  const float* l0_bih    = (const float*)d_in[4];
  const float* l0_bhh    = (const float*)d_in[5];
  const float* l1_Wih    = (const float*)d_in[6];
  const float* l1_Whh    = (const float*)d_in[7];
  const float* l1_bih    = (const float*)d_in[8];
  const float* l1_bhh    = (const float*)d_in[9];
  const float* fc_W      = (const float*)d_in[10];
  const float* fc_b      = (const float*)d_in[11];
  const float* conv1_W   = (const float*)d_in[12];
  const float* conv1_b   = (const float*)d_in[13];
  const float* pconv_W   = (const float*)d_in[14];   // [256,256,5,5]
  const float* pconv_b   = (const float*)d_in[15];   // [256]
  const float* dcap_W    = (const float*)d_in[16];   // [12800,10,16,8]
  float* out = (float*)d_out;

  float* ws = (float*)d_ws;
  float* enc    = ws;                    // 122880
  float* h0     = enc    + 122880;       // 245760
  float* h1     = h0     + 245760;       // 245760
  float* xg     = h1     + 245760;       // 983040 (reused by both layers)
  float* incap  = xg     + 983040;       // 14400
  float* c1out  = incap  + 14400;        // 2359296
  float* y      = c1out  + 2359296;      // 1638400
  float* scale  = y      + 1638400;      // 128
  float* umat   = scale  + 128;          // 1638400
  float* sbuf   = umat   + 1638400;      // 2560

  k_embed<<<480, 256, 0, stream>>>(input, embed, enc);
  k_xg<<<dim3(30, 8, 2), 256, 0, stream>>>(enc, l0_Wih, l0_bih, l0_bhh, xg, 256);
  k_lstm_rec<<<2, 512, 0, stream>>>(xg, l0_Whh, h0);
  k_xg<<<dim3(30, 8, 2), 256, 0, stream>>>(h0, l1_Wih, l1_bih, l1_bhh, xg, 512);
  k_lstm_rec<<<2, 512, 0, stream>>>(xg, l1_Whh, h1);
  k_fc<<<57, 256, 0, stream>>>(h1, fc_W, fc_b, incap);
  k_conv1<<<9216, 256, 0, stream>>>(incap, conv1_W, conv1_b, c1out);
  k_pconv<<<dim3(400, 2), 256, 0, stream>>>(c1out, pconv_W, pconv_b, y);
  k_scale<<<dim3(16, 8), 256, 0, stream>>>(y, scale);
  k_umat<<<6400, 256, 0, stream>>>(y, scale, umat);
  k_szero<<<10, 256, 0, stream>>>(sbuf);
  k_sgemm<<<64, 320, 0, stream>>>(umat, dcap_W, sbuf);
  k_vout<<<1, 256, 0, stream>>>(sbuf, out);
  k_feats<<<32, 256, 0, stream>>>(h1, out);
}